// HashEmbeddings_12567074308161
// MI455X (gfx1250) — compile-verified
//
#include <hip/hip_runtime.h>

typedef float v2f __attribute__((ext_vector_type(2)));
typedef float v4f __attribute__((ext_vector_type(4)));
typedef float v8f __attribute__((ext_vector_type(8)));
typedef unsigned int u32x4 __attribute__((ext_vector_type(4)));
typedef int i32x4 __attribute__((ext_vector_type(4)));
typedef int i32x8 __attribute__((ext_vector_type(8)));

#define BATCH    2048
#define BUCKETS  100000
#define DIM      64
#define KT       128                          // K panel staged in LDS
#define NPANELS  ((BUCKETS + KT - 1) / KT)    // 782 (tail zero-filled by TDM OOB)
#define ASTR     (KT + 4)                     // 132 floats: 4-bank rotate per row
#define BSTR     (DIM + 8)                    // 72 floats: 16-bank shift per 2 rows
#define A_FLOATS (16 * ASTR)                  // 2112
#define B_FLOATS (KT * BSTR)                  // 9216

// dynamic LDS layout (float offsets) — double-buffered panels + epilogue
#define OFF_A0   0
#define OFF_A1   (OFF_A0 + A_FLOATS)
#define OFF_B0   (OFF_A1 + A_FLOATS)
#define OFF_B1   (OFF_B0 + B_FLOATS)
#define OFF_RED  (OFF_B1 + B_FLOATS)          // 4*3*8*32 = 3072
#define OFF_RSUM (OFF_RED + 4 * 3 * 8 * 32)   // 16*32 = 512
#define OFF_INV  (OFF_RSUM + 16 * 32)         // 16
#define SMEM_FLOATS (OFF_INV + 16)
#define SMEM_BYTES  (SMEM_FLOATS * 4)         // 105,024 B  (< 320 KB/WGP)

#define NTHREADS 512                          // 16 waves: 4 n-tiles x 4 k-quarters

#if __has_builtin(__builtin_amdgcn_tensor_load_to_lds) && \
    __has_builtin(__builtin_amdgcn_s_wait_tensorcnt)
#define HAVE_TDM 1
#else
#define HAVE_TDM 0
#endif

__launch_bounds__(NTHREADS, 1)
__global__ void hash_embed_mean_gemm(const float* __restrict__ inp,
                                     const float* __restrict__ emb,
                                     float* __restrict__ out) {
  extern __shared__ float smem[];
  float* const RedB = smem + OFF_RED;
  float* const Rsum = smem + OFF_RSUM;
  float* const Inv  = smem + OFF_INV;

  const int tid   = threadIdx.x;
  const int lane  = tid & 31;
  const int wave  = tid >> 5;
  const int ntile = wave & 3;                 // 16-wide column tile of DIM
  const int kq    = wave >> 2;                // quarter of the K panel
  const int mblk  = blockIdx.x * 16;

  const int half  = lane >> 4;                // lanes 16..31 carry K+2,K+3
  const int mrow  = lane & 15;
  const int arow  = tid >> 5;                 // 0..15: M row owned for row-sums
  const int acol  = tid & 31;                 // float4 slot within panel row

  float rsum = 0.0f;
  v8f acc = {};

  // ---- WMMA consumer: V_WMMA_F32_16X16X4_F32 over one K panel quarter ----
  auto compute_panel = [&](const float* As, const float* Bs) {
    const int kbase = kq * (KT / 4);
#pragma unroll
    for (int kk = 0; kk < KT / 4; kk += 4) {
      const int k = kbase + kk + 2 * half;
      v2f a, b;
      // A 16x4 f32: VGPR0 = K0 (lanes 0-15) / K2 (lanes 16-31), VGPR1 = K1/K3
      a.x = As[mrow * ASTR + k + 0];
      a.y = As[mrow * ASTR + k + 1];
      // B 4x16 f32 mirrors A across K
      b.x = Bs[(k + 0) * BSTR + ntile * 16 + mrow];
      b.y = Bs[(k + 1) * BSTR + ntile * 16 + mrow];
      acc = __builtin_amdgcn_wmma_f32_16x16x4_f32(false, a, false, b,
                                                  (short)0, acc, false, false);
    }
  };

#if HAVE_TDM
  // ---- Tensor Data Mover producer: one D# per matrix panel, wave 0 only ----
  auto tdm_issue = [&](int buf, int k0) {
    if (wave != 0) return;
    const unsigned kv = (unsigned)((BUCKETS - k0) < KT ? (BUCKETS - k0) : KT);
    i32x4 z4 = {};
    // A tile: 16 rows x KT dwords, row stride BUCKETS dwords, pad 4 dw / 128 dw
    {
      const unsigned long long ga =
          (unsigned long long)(size_t)(inp + (size_t)mblk * BUCKETS + k0);
      const unsigned lds = (unsigned)(size_t)(smem + (buf ? OFF_A1 : OFF_A0));
      u32x4 g0;
      g0.x = 1u;                                   // count=1, user descriptor
      g0.y = lds;                                  // lds_addr
      g0.z = (unsigned)ga;                         // global_addr[31:0]
      g0.w = (unsigned)((ga >> 32) & 0x01ffffffull) | (2u << 30);  // type=2
      i32x8 g1;
      g1[0] = (int)((2u << 16) |                   // data_size = 4 B
                    (1u << 20) |                   // pad_enable
                    (6u << 22) |                   // pad_interval: 128 dwords
                    (3u << 25));                   // pad_amount:   4 dwords
      g1[1] = (int)((kv & 0xffffu) << 16);         // tensor_dim0[15:0] (valid K)
      g1[2] = (int)(((kv >> 16) & 0xffffu) | (16u << 16));  // td0 hi | td1=16 rows
      g1[3] = (int)(((unsigned)KT & 0xffffu) << 16);        // tile_dim0 = KT
      g1[4] = 16;                                  // tile_dim1 = 16 rows
      g1[5] = BUCKETS;                             // tensor_dim0_stride lo
      g1[6] = 0;
      g1[7] = 0;
#if __clang_major__ >= 23
      i32x8 z8 = {};
      __builtin_amdgcn_tensor_load_to_lds(g0, g1, z4, z4, z8, 0);
#else
      __builtin_amdgcn_tensor_load_to_lds(g0, g1, z4, z4, 0);
#endif
    }
    // B tile: kv rows x DIM dwords, row stride DIM dwords, pad 8 dw / 64 dw
    {
      const unsigned long long ga =
          (unsigned long long)(size_t)(emb + (size_t)k0 * DIM);
      const unsigned lds = (unsigned)(size_t)(smem + (buf ? OFF_B1 : OFF_B0));
      u32x4 g0;
      g0.x = 1u;
      g0.y = lds;
      g0.z = (unsigned)ga;
      g0.w = (unsigned)((ga >> 32) & 0x01ffffffull) | (2u << 30);
      i32x8 g1;
      g1[0] = (int)((2u << 16) |                   // data_size = 4 B
                    (1u << 20) |                   // pad_enable
                    (5u << 22) |                   // pad_interval: 64 dwords
                    (7u << 25));                   // pad_amount:   8 dwords
      g1[1] = (int)(((unsigned)DIM & 0xffffu) << 16);       // tensor_dim0 = 64
      g1[2] = (int)((kv & 0xffffu) << 16);                  // tensor_dim1 = valid rows
      g1[3] = (int)(((kv >> 16) & 0xffffu) |
                    (((unsigned)DIM & 0xffffu) << 16));     // tile_dim0 = 64
      g1[4] = KT;                                  // tile_dim1 = 128 rows (OOB rows -> 0)
      g1[5] = DIM;                                 // tensor_dim0_stride
      g1[6] = 0;
      g1[7] = 0;
#if __clang_major__ >= 23
      i32x8 z8 = {};
      __builtin_amdgcn_tensor_load_to_lds(g0, g1, z4, z4, z8, 0);
#else
      __builtin_amdgcn_tensor_load_to_lds(g0, g1, z4, z4, 0);
#endif
    }
  };

  // ---- pipeline: TDM(p+1 -> other buffer) overlaps WMMA(p); 1 barrier/panel ----
  tdm_issue(0, 0);
  if (wave == 0) __builtin_amdgcn_s_wait_tensorcnt(0);
  __syncthreads();

  for (int p = 0; p < NPANELS; ++p) {
    const int cur = p & 1;
    const float* As = smem + (cur ? OFF_A1 : OFF_A0);
    const float* Bs = smem + (cur ? OFF_B1 : OFF_B0);
    if (p + 1 < NPANELS) tdm_issue(1 - cur, (p + 1) * KT);
    compute_panel(As, Bs);
    {  // row-sum side-band: each thread re-reads its owned A row chunk
      const v4f av = *(const v4f*)(As + arow * ASTR + acol * 4);
      rsum += av.x + av.y + av.z + av.w;
    }
    if (wave == 0 && p + 1 < NPANELS) __builtin_amdgcn_s_wait_tensorcnt(0);
    __syncthreads();
  }

#else  // !HAVE_TDM — fallback: reg-staged global->LDS double buffering
  v4f aReg;
  v4f bReg[4];
  const float* arowp = inp + (size_t)(mblk + arow) * BUCKETS;

  auto load_regs = [&](int k0) {
    const int ka = k0 + acol * 4;
    if (ka < BUCKETS) aReg = *(const v4f*)(arowp + ka);
    else              aReg = v4f{0.f, 0.f, 0.f, 0.f};
#pragma unroll
    for (int i = 0; i < 4; ++i) {
      const int f  = tid + i * NTHREADS;
      const int kr = f >> 4;
      const int j4 = f & 15;
      const int kb = k0 + kr;
      if (kb < BUCKETS) bReg[i] = *(const v4f*)(emb + (size_t)kb * DIM + j4 * 4);
      else              bReg[i] = v4f{0.f, 0.f, 0.f, 0.f};
    }
  };
  auto store_regs = [&](int buf) {
    float* As = smem + (buf ? OFF_A1 : OFF_A0);
    float* Bs = smem + (buf ? OFF_B1 : OFF_B0);
    *(v4f*)(As + arow * ASTR + acol * 4) = aReg;
    rsum += aReg.x + aReg.y + aReg.z + aReg.w;
#pragma unroll
    for (int i = 0; i < 4; ++i) {
      const int f  = tid + i * NTHREADS;
      const int kr = f >> 4;
      const int j4 = f & 15;
      *(v4f*)(Bs + kr * BSTR + j4 * 4) = bReg[i];
    }
  };

  load_regs(0);
  store_regs(0);
  __syncthreads();
  for (int p = 0; p < NPANELS; ++p) {
    const int cur = p & 1;
    if (p + 1 < NPANELS) load_regs((p + 1) * KT);
    compute_panel(smem + (cur ? OFF_A1 : OFF_A0), smem + (cur ? OFF_B1 : OFF_B0));
    if (p + 1 < NPANELS) store_regs(1 - cur);
    __syncthreads();
  }
#endif

  // ---- epilogue: row-sum reduce, 1/s (0 if empty), cross-quarter reduce ----
  Rsum[arow * 32 + acol] = rsum;
  __syncthreads();
  if (tid < 16) {
    float s = 0.f;
#pragma unroll
    for (int j = 0; j < 32; ++j) s += Rsum[tid * 32 + j];
    Inv[tid] = (s > 0.f) ? (1.0f / s) : 0.0f;
  }
  if (kq != 0) {
#pragma unroll
    for (int v = 0; v < 8; ++v)
      RedB[(((ntile * 3) + (kq - 1)) * 8 + v) * 32 + lane] = acc[v];
  }
  __syncthreads();

  if (kq == 0) {
#pragma unroll
    for (int v = 0; v < 8; ++v) {
      float c = acc[v];
#pragma unroll
      for (int q = 0; q < 3; ++q)
        c += RedB[(((ntile * 3) + q) * 8 + v) * 32 + lane];
      // C/D layout: VGPR v, lane l -> M = v + 8*(l>>4), N = l&15
      const int m = v + 8 * half;
      out[(size_t)(mblk + m) * DIM + ntile * 16 + mrow] = c * Inv[m];
    }
  }
}

extern "C" void kernel_launch(void* const* d_in, const int* in_sizes, int n_in,
                              void* d_out, int out_size, void* d_ws, size_t ws_size,
                              hipStream_t stream) {
  (void)in_sizes; (void)n_in; (void)out_size; (void)d_ws; (void)ws_size;
  // raise dynamic-LDS cap (unconditional + idempotent -> capture/deterministic safe)
  (void)hipFuncSetAttribute(reinterpret_cast<const void*>(&hash_embed_mean_gemm),
                            hipFuncAttributeMaxDynamicSharedMemorySize, SMEM_BYTES);
  const float* inp = (const float*)d_in[0];   // [2048, 100000] f32
  const float* emb = (const float*)d_in[1];   // [100000, 64]  f32
  float*       out = (float*)d_out;           // [2048, 64]    f32
  hash_embed_mean_gemm<<<dim3(BATCH / 16), dim3(NTHREADS), SMEM_BYTES, stream>>>(inp, emb, out);
}